// DiffusionGraphConv_57088705298510
// MI455X (gfx1250) — compile-verified
//
#include <hip/hip_runtime.h>

// ---------------------------------------------------------------------------
// DiffusionGraphConv on MI455X (gfx1250, wave32)
//   N=2048 nodes, F=128 (input 64 + hidden 64), B=64 batch, M=5 matrices,
//   OUT=64.  Dominant cost: 4 fp32 GEMMs [2048x2048] @ [2048x8192].
//   fp32 precision -> V_WMMA_F32_16X16X4_F32.
//   Tile staging uses GLOBAL_LOAD_ASYNC_TO_LDS_B128 (ASYNCcnt) with
//   double-buffered LDS so the async DMA overlaps the WMMA compute.
// ---------------------------------------------------------------------------

typedef float v2f __attribute__((ext_vector_type(2)));
typedef float v8f __attribute__((ext_vector_type(8)));

#define NN   2048          // nodes
#define FB   8192          // F * B = 128 * 64
#define BB   64            // batch
#define KTOT 2048          // K of the big GEMMs
#define KW   640           // F * NUM_MATRICES for the final contraction
#define OUTD 64

#define BM 128
#define BN 128
#define BK 32
#define ASTR 36            // padded LDS stride for S tile rows (32 + 4), 144 B
#define XSTR 136           // padded LDS stride for X tile rows (128 + 8), 544 B

static __device__ __forceinline__ v8f wmma4(v2f a, v2f b, v8f c) {
    // D(16x16,f32) = A(16x4,f32) * B(4x16,f32) + C
    return __builtin_amdgcn_wmma_f32_16x16x4_f32(
        /*neg_a=*/false, a, /*neg_b=*/false, b,
        /*c_mod=*/(short)0, c, /*reuse_a=*/false, /*reuse_b=*/false);
}

// Async 16-byte global -> LDS copy (per lane).  Data bypasses VGPRs entirely;
// completion tracked by ASYNCcnt.
static __device__ __forceinline__ void async_cp16(unsigned lds_off,
                                                  const float* src) {
    asm volatile("global_load_async_to_lds_b128 %0, %1, off"
                 :: "v"(lds_off), "v"((unsigned long long)(size_t)src)
                 : "memory");
}

static __device__ __forceinline__ void wait_async0() {
    asm volatile("s_wait_asynccnt 0" ::: "memory");
}

// Issue the async copies for one (S tile 128x32, X tile 32x128) pair.
static __device__ __forceinline__ void stage_tiles(const float* __restrict__ S,
                                                   const float* __restrict__ X,
                                                   float* AsBuf, float* BsBuf,
                                                   int tid, int rowBase,
                                                   int colBase, int kt) {
#pragma unroll
    for (int i = 0; i < 4; ++i) {
        int v  = tid + i * 256;           // float4 slot 0..1023
        int r  = v >> 3;                  // 8 float4 per 32-wide row
        int c4 = (v & 7) << 2;
        async_cp16((unsigned)(size_t)&AsBuf[r * ASTR + c4],
                   S + (size_t)(rowBase + r) * KTOT + kt + c4);
    }
#pragma unroll
    for (int i = 0; i < 4; ++i) {
        int v  = tid + i * 256;
        int r  = v >> 5;                  // 32 float4 per 128-wide row
        int c4 = (v & 31) << 2;
        async_cp16((unsigned)(size_t)&BsBuf[r * XSTR + c4],
                   X + (size_t)(kt + r) * FB + colBase + c4);
    }
}

// ---------------------------------------------------------------------------
// Kernel 1: build x0[n, f*B + b] from inputs[b, n*64+f] / state[b, n*64+(f-64)]
// ---------------------------------------------------------------------------
__global__ __launch_bounds__(256)
void pack_x0_kernel(const float* __restrict__ inp, const float* __restrict__ st,
                    float* __restrict__ x0) {
    size_t idx = (size_t)blockIdx.x * 256 + threadIdx.x;   // over N*F*B
    int b = (int)(idx & 63);
    int f = (int)((idx >> 6) & 127);
    int n = (int)(idx >> 13);
    size_t src = (size_t)b * (NN * 64) + (size_t)n * 64;
    float v = (f < 64) ? inp[src + f] : st[src + (f - 64)];
    x0[idx] = v;
}

// ---------------------------------------------------------------------------
// Kernel 2: Y = alpha * (S @ X) + beta * Z   (S: [2048,2048], X/Z/Y: [2048,8192])
//   block tile 128x128, 256 threads = 8 waves (2 along M x 4 along N),
//   wave tile 64x32 = 4x2 fragments of 16x16 fp32 accumulators.
//   Double-buffered LDS, filled by async global->LDS DMA.
// ---------------------------------------------------------------------------
__global__ __launch_bounds__(256)
void gemm_wmma_kernel(const float* __restrict__ S, const float* __restrict__ X,
                      const float* __restrict__ Z, float* __restrict__ Y,
                      float alpha, float beta, int useZ) {
    __shared__ __align__(16) float As[2][BM * ASTR];   // 2 x 18 KB
    __shared__ __align__(16) float Bs[2][BK * XSTR];   // 2 x 17 KB

    const int tid  = threadIdx.x;
    const int lane = tid & 31;
    const int wave = tid >> 5;        // 0..7
    const int wm   = wave & 1;        // wave row (M)
    const int wn   = wave >> 1;       // wave col (N)
    const int l16  = lane & 15;
    const int lh   = lane >> 4;       // lane half selects K pair / M+8

    const int rowBase = blockIdx.y * BM;
    const int colBase = blockIdx.x * BN;

    v8f acc[4][2];
#pragma unroll
    for (int mi = 0; mi < 4; ++mi)
#pragma unroll
        for (int ni = 0; ni < 2; ++ni)
            acc[mi][ni] = (v8f){0.f, 0.f, 0.f, 0.f, 0.f, 0.f, 0.f, 0.f};

    // prologue: stage tile 0 into buffer 0
    stage_tiles(S, X, As[0], Bs[0], tid, rowBase, colBase, 0);
    wait_async0();
    __syncthreads();

    int buf = 0;
    for (int kt = 0; kt < KTOT; kt += BK, buf ^= 1) {
        // kick off the DMA for the next tile into the other buffer
        if (kt + BK < KTOT)
            stage_tiles(S, X, As[buf ^ 1], Bs[buf ^ 1], tid, rowBase, colBase,
                        kt + BK);

        const float* Asb = As[buf];
        const float* Bsb = Bs[buf];
#pragma unroll
        for (int kk = 0; kk < BK; kk += 4) {
            const int ka = kk + lh * 2;       // this half-wave's K pair
            v2f a[4], b[2];
#pragma unroll
            for (int mi = 0; mi < 4; ++mi) {
                int r = wm * 64 + mi * 16 + l16;
                a[mi] = *(const v2f*)&Asb[r * ASTR + ka];     // ds_load_b64
            }
#pragma unroll
            for (int ni = 0; ni < 2; ++ni) {
                int c = wn * 32 + ni * 16 + l16;
                v2f t;
                t[0] = Bsb[ka * XSTR + c];
                t[1] = Bsb[(ka + 1) * XSTR + c];
                b[ni] = t;
            }
#pragma unroll
            for (int mi = 0; mi < 4; ++mi)
#pragma unroll
                for (int ni = 0; ni < 2; ++ni)
                    acc[mi][ni] = wmma4(a[mi], b[ni], acc[mi][ni]);
        }

        // publish the prefetched tile & fence the buffer we just read
        wait_async0();
        __syncthreads();
    }

    // ---- epilogue: Y = alpha*acc (+ beta*Z) ----
#pragma unroll
    for (int mi = 0; mi < 4; ++mi) {
#pragma unroll
        for (int ni = 0; ni < 2; ++ni) {
            int col = colBase + wn * 32 + ni * 16 + l16;
#pragma unroll
            for (int r = 0; r < 8; ++r) {
                int row = rowBase + wm * 64 + mi * 16 + lh * 8 + r;
                size_t idx = (size_t)row * FB + col;
                float v = alpha * acc[mi][ni][r];
                if (useZ) v += beta * Z[idx];
                Y[idx] = v;
            }
        }
    }
}

// ---------------------------------------------------------------------------
// Kernel 3: per-node GEMM  out[b, n*64+o] = bias[o] +
//             sum_{k=0..639} A_n[b][k] * W[k][o],  k = f*5+m,
//             A_n[b][k] = mats[m][n*8192 + f*64 + b]
//   One block per node, 128 threads = 4 waves; wave w owns o in [16w,16w+16).
// ---------------------------------------------------------------------------
__global__ __launch_bounds__(128)
void final_wmma_kernel(const float* __restrict__ mats,   // 5 matrices, each N*FB
                       const float* __restrict__ W,      // [640, 64]
                       const float* __restrict__ bias,   // [64]
                       float* __restrict__ out) {
    const int n    = blockIdx.x;
    const int lane = threadIdx.x & 31;
    const int wave = threadIdx.x >> 5;    // 0..3 -> output column strip
    const int l16  = lane & 15;
    const int lh   = lane >> 4;
    const size_t MSZ = (size_t)NN * FB;   // elements per stored matrix
    const size_t nbase = (size_t)n * FB;

    v8f acc[4];
#pragma unroll
    for (int mi = 0; mi < 4; ++mi)
        acc[mi] = (v8f){0.f, 0.f, 0.f, 0.f, 0.f, 0.f, 0.f, 0.f};

    const int ocol = wave * 16 + l16;

    for (int k = 0; k < KW; k += 4) {
        const int k0 = k + lh * 2;
        // B fragment from the weight matrix (row-major [640,64])
        v2f b;
        b[0] = W[(size_t)k0 * OUTD + ocol];
        b[1] = W[(size_t)(k0 + 1) * OUTD + ocol];
        // decompose k -> (f, m) for the two K slots of this half-wave
        const int f0 = k0 / 5, m0 = k0 - 5 * f0;
        const int f1 = (k0 + 1) / 5, m1 = (k0 + 1) - 5 * f1;
        const size_t a0 = (size_t)m0 * MSZ + nbase + (size_t)f0 * BB;
        const size_t a1 = (size_t)m1 * MSZ + nbase + (size_t)f1 * BB;
#pragma unroll
        for (int mi = 0; mi < 4; ++mi) {
            int bb = mi * 16 + l16;           // batch row (M dim)
            v2f a;
            a[0] = mats[a0 + bb];             // consecutive lanes -> consecutive addrs
            a[1] = mats[a1 + bb];
            acc[mi] = wmma4(a, b, acc[mi]);
        }
    }

#pragma unroll
    for (int mi = 0; mi < 4; ++mi) {
#pragma unroll
        for (int r = 0; r < 8; ++r) {
            int brow = mi * 16 + lh * 8 + r;  // batch index
            out[(size_t)brow * (NN * OUTD) + (size_t)n * OUTD + ocol] =
                acc[mi][r] + bias[ocol];
        }
    }
}

// ---------------------------------------------------------------------------
extern "C" void kernel_launch(void* const* d_in, const int* in_sizes, int n_in,
                              void* d_out, int out_size, void* d_ws, size_t ws_size,
                              hipStream_t stream) {
    const float* s0     = (const float*)d_in[0];
    const float* s1     = (const float*)d_in[1];
    const float* inputs = (const float*)d_in[2];
    const float* state  = (const float*)d_in[3];
    const float* weight = (const float*)d_in[4];
    const float* biases = (const float*)d_in[5];
    float* out = (float*)d_out;

    const size_t MSZ = (size_t)NN * FB;       // 16.7M floats per matrix
    float* ws    = (float*)d_ws;
    float* x0    = ws + 0 * MSZ;              // mats[0]
    float* x1_s0 = ws + 1 * MSZ;              // mats[1]
    float* x2_s0 = ws + 2 * MSZ;              // mats[2]
    float* x1_s1 = ws + 3 * MSZ;              // mats[3]
    float* x2_s1 = ws + 4 * MSZ;              // mats[4]

    // 1) pack x0
    pack_x0_kernel<<<dim3((unsigned)(MSZ / 256)), dim3(256), 0, stream>>>(
        inputs, state, x0);

    dim3 ggrid(FB / BN, NN / BM);             // 64 x 16
    dim3 gblk(256);

    // 2) x1 = s0 @ x0
    gemm_wmma_kernel<<<ggrid, gblk, 0, stream>>>(s0, x0, x0, x1_s0, 1.0f, 0.0f, 0);
    // 3) x2 = 2*(s0 @ x1) - x0
    gemm_wmma_kernel<<<ggrid, gblk, 0, stream>>>(s0, x1_s0, x0, x2_s0, 2.0f, -1.0f, 1);
    // 4) x1' = s1 @ x0
    gemm_wmma_kernel<<<ggrid, gblk, 0, stream>>>(s1, x0, x0, x1_s1, 1.0f, 0.0f, 0);
    // 5) x2' = 2*(s1 @ x1') - x0
    gemm_wmma_kernel<<<ggrid, gblk, 0, stream>>>(s1, x1_s1, x0, x2_s1, 2.0f, -1.0f, 1);

    // 6) out = concat-features @ weight + bias
    final_wmma_kernel<<<dim3(NN), dim3(128), 0, stream>>>(ws, weight, biases, out);
}